// BioPhysNet_31791347925420
// MI455X (gfx1250) — compile-verified
//
#include <hip/hip_runtime.h>
#include <hip/hip_bf16.h>
#include <math.h>

// ---------------------------------------------------------------------------
// BioPhysNet forward for MI455X (gfx1250, wave32).
//  Kernel 1: per-sample algebra -> 128-float param block in d_ws.
//  Kernel 2: one wave32 per 16-pixel tile; 33-ch bilinear gather into an LDS
//            B-panel, then a 9-op V_WMMA_F32_16X16X4_F32 chain computes the
//            3x33 spectral contraction for 16 pixels at once.
// ---------------------------------------------------------------------------

typedef __attribute__((ext_vector_type(2))) float v2f;
typedef __attribute__((ext_vector_type(8))) float v8f;

#define NSAMP   256
#define HW      4096            // 64*64
#define PSTRIDE 128             // floats per sample param block

// output tuple offsets (floats), in reference return order
#define OFF_RGB   0
#define OFF_SHADE 3145728       // 256*3*4096
#define OFF_SPEC  4194304
#define OFF_BLOOD 7340032
#define OFF_MEL   8388608
#define OFF_B     9437184
#define OFF_RAW   9437696

__device__ __forceinline__ float sigmoidf(float x) {
    return 1.0f / (1.0f + __expf(-x));
}

// ---------------------------------------------------------------------------
// Kernel 1: per-sample prep (256 threads total, one sample each)
// param block layout:  [0..107]  eS[c*36+k]  (= e[k]*S[c][k], k>=33 -> 0)
//                      [108..110] lightColor[c]
//                      [111..119] M'[r*3+j]  (= (T_rgb . ts^T)[r][j]/LC[j])
//                      [120..121] b0, b1
// ---------------------------------------------------------------------------
__global__ __launch_bounds__(32)
void bpn_prep(const float* __restrict__ lighting, const float* __restrict__ illA,
              const float* __restrict__ illF,     const float* __restrict__ pcaMeans,
              const float* __restrict__ pcaComp,  const float* __restrict__ tmatrix,
              const float* __restrict__ s0,       const float* __restrict__ s1,
              const float* __restrict__ s2,       const float* __restrict__ t2rgb,
              float* __restrict__ params,         float* __restrict__ out)
{
    int n = blockIdx.x * 32 + threadIdx.x;
    if (n >= NSAMP) return;
    const float* lp = lighting + n * 17;

    // softmax over first 14
    float mx = lp[0];
    #pragma unroll
    for (int i = 1; i < 14; ++i) mx = fmaxf(mx, lp[i]);
    float ex[14], se = 0.f;
    #pragma unroll
    for (int i = 0; i < 14; ++i) { ex[i] = __expf(lp[i] - mx); se += ex[i]; }
    float inv_se = 1.f / se;
    float wA = ex[0] * inv_se, wD = ex[1] * inv_se;
    float fW[12];
    #pragma unroll
    for (int f = 0; f < 12; ++f) fW[f] = ex[2 + f] * inv_se;

    float colorTemp = sigmoidf(lp[14]);
    float b0 = 6.f * sigmoidf(lp[15]) - 3.f;
    float b1 = 6.f * sigmoidf(lp[16]) - 3.f;

    // blackbody chromaticity
    float t  = colorTemp * 21000.f + 4000.f;
    float it = 1.f / t, it2 = it * it, it3 = it2 * it;
    float x1 = -4.607e9f * it3 + 2.9678e6f * it2 + 99.11f * it + 0.244063f;
    float x2 = -2.0064e9f * it3 + 1.9018e6f * it2 + 247.48f * it + 0.23704f;
    float x  = (t <= 7000.f) ? x1 : x2;
    float y  = -3.f * x * x + 2.87f * x - 0.275f;
    float mm = 0.0241f + 0.2562f * x - 0.7341f * y;
    float m1 = (-1.3515f - 1.7703f * x + 5.9114f * y) / mm;
    float m2 = (0.03f - 31.4424f * x + 30.0717f * y) / mm;

    // sum of daylight spectrum s
    float ssum = 0.f;
    for (int k = 0; k < 33; ++k) ssum += s0[k] + m1 * s1[k] + m2 * s2[k];
    float inv_ssum = 1.f / ssum;

    // pass 1: sum of unnormalized e
    float esum = 0.f;
    for (int k = 0; k < 33; ++k) {
        float sk = s0[k] + m1 * s1[k] + m2 * s2[k];
        float er = wA * illA[k] + wD * sk * inv_ssum;
        #pragma unroll
        for (int f = 0; f < 12; ++f) er += fW[f] * illF[k * 12 + f];
        esum += er;
    }
    float inv_esum = 1.f / esum;

    // pass 2: S, lightColor, eS into param block
    float* P = params + n * PSTRIDE;
    float LC0 = 0.f, LC1 = 0.f, LC2 = 0.f;
    for (int k = 0; k < 33; ++k) {
        float sk = s0[k] + m1 * s1[k] + m2 * s2[k];
        float er = wA * illA[k] + wD * sk * inv_ssum;
        #pragma unroll
        for (int f = 0; f < 12; ++f) er += fW[f] * illF[k * 12 + f];
        er *= inv_esum;
        #pragma unroll
        for (int c = 0; c < 3; ++c) {
            int j = c * 33 + k;
            float v = fmaxf(b0 * pcaComp[j * 2] + b1 * pcaComp[j * 2 + 1] + pcaMeans[j], 0.f);
            if (c == 0) LC0 += v; else if (c == 1) LC1 += v; else LC2 += v;
            P[c * 36 + k] = v * er;
        }
    }
    #pragma unroll
    for (int c = 0; c < 3; ++c) { P[c*36+33] = 0.f; P[c*36+34] = 0.f; P[c*36+35] = 0.f; }
    P[108] = LC0; P[109] = LC1; P[110] = LC2;

    // ts: bilinear sample of tmatrix (9,128,128) at (b0/3, b1/3)
    float gx = (b0 * (1.f/3.f) + 1.f) * 0.5f * 127.f;
    float gy = (b1 * (1.f/3.f) + 1.f) * 0.5f * 127.f;
    float fx0 = floorf(gx), fy0 = floorf(gy);
    float twx = gx - fx0, twy = gy - fy0;
    int x0i = min(max((int)fx0, 0), 127), xi1 = min(x0i + 1, 127);
    int y0i = min(max((int)fy0, 0), 127), yi1 = min(y0i + 1, 127);
    float w00 = (1.f-twy)*(1.f-twx), w01 = (1.f-twy)*twx;
    float w10 = twy*(1.f-twx),       w11 = twy*twx;
    float tsf[9];
    #pragma unroll
    for (int j = 0; j < 9; ++j) {
        const float* img = tmatrix + j * 16384;
        tsf[j] = w00 * img[y0i*128 + x0i] + w01 * img[y0i*128 + xi1]
               + w10 * img[yi1*128 + x0i] + w11 * img[yi1*128 + xi1];
    }
    // M'[r][j] = (sum_c T[r][c]*ts[j][c]) / LC[j]
    float invLC[3] = {1.f/LC0, 1.f/LC1, 1.f/LC2};
    #pragma unroll
    for (int r = 0; r < 3; ++r)
        #pragma unroll
        for (int j = 0; j < 3; ++j) {
            float acc = 0.f;
            #pragma unroll
            for (int c = 0; c < 3; ++c) acc += t2rgb[r*3 + c] * tsf[j*3 + c];
            P[111 + r*3 + j] = acc * invLC[j];
        }
    P[120] = b0; P[121] = b1;

    out[OFF_B + n*2 + 0] = b0;
    out[OFF_B + n*2 + 1] = b1;
}

// ---------------------------------------------------------------------------
// Kernel 2: one wave32 block per 16-pixel tile.
//  Fragment layout assumptions (CDNA5 ISA 7.12.2, f32 16x16x4):
//   A (16x4, M on lanes): lane l -> M=l&15; VGPR0 holds K=2*(l>>4), VGPR1 K+1
//   B (4x16, N on lanes): lane l -> N=l&15; VGPR0 holds K=2*(l>>4), VGPR1 K+1
//   C/D (16x16):          lane l -> N=l&15; VGPR v holds M = v + 8*(l>>4)
// ---------------------------------------------------------------------------
__global__ __launch_bounds__(32)
void bpn_pixel(const float* __restrict__ mel,   const float* __restrict__ blood,
               const float* __restrict__ shade, const float* __restrict__ spec,
               const float* __restrict__ skinColor, const float* __restrict__ params,
               float* __restrict__ out)
{
    const int lane  = threadIdx.x;
    const int n     = blockIdx.x >> 8;     // 256 tiles per sample
    const int tile  = blockIdx.x & 255;
    const int pbase = tile * 16;
    const int hi    = lane >> 4;
    const int pi    = lane & 15;

    __shared__ float sPA[16 * 36];         // padded A panel: rows 3..15 zero
    __shared__ float sMeta[32];            // params[96..127]: LC, M', b
    __shared__ float sW[64];               // 16 pixels x 4 bilinear weights
    __shared__ int   sA[64];               // 16 pixels x 4 corner offsets
    __shared__ float sB[36 * 16];          // B panel: K x N (K padded to 36)

    // stage A panel (zero-padded) + metadata into LDS, coalesced
    const float* P = params + n * PSTRIDE;
    #pragma unroll
    for (int i = 0; i < 18; ++i) {
        int idx = i * 32 + lane;           // 576 = 16*36 entries
        float v = 0.f;
        if (idx < 108) v = P[idx];
        sPA[idx] = v;
    }
    sMeta[lane] = P[96 + lane];            // sMeta[12..14]=LC, [15..23]=M'

    float m_v = 0.f, bl_v = 0.f, sh_v = 0.f, sp_v = 0.f;
    if (lane < 16) {
        int gi = n * HW + pbase + lane;
        m_v  = 2.f * sigmoidf(mel[gi])   - 1.f;
        bl_v = 2.f * sigmoidf(blood[gi]) - 1.f;
        sh_v = __expf(shade[gi]);
        sp_v = __expf(spec[gi]);
        float gx = (m_v  + 1.f) * 0.5f * 255.f;   // grid ch0 = mel -> x
        float gy = (bl_v + 1.f) * 0.5f * 255.f;   // grid ch1 = blood -> y
        float fx0 = floorf(gx), fy0 = floorf(gy);
        float wx = gx - fx0, wy = gy - fy0;
        int x0i = min(max((int)fx0, 0), 255), x1i = min(x0i + 1, 255);
        int y0i = min(max((int)fy0, 0), 255), y1i = min(y0i + 1, 255);
        sW[lane*4+0] = (1.f-wy)*(1.f-wx);  sA[lane*4+0] = y0i*256 + x0i;
        sW[lane*4+1] = (1.f-wy)*wx;        sA[lane*4+1] = y0i*256 + x1i;
        sW[lane*4+2] = wy*(1.f-wx);        sA[lane*4+2] = y1i*256 + x0i;
        sW[lane*4+3] = wy*wx;              sA[lane*4+3] = y1i*256 + x1i;
    }
    __syncthreads();   // single-wave WG: degenerates to S_NOP + dscnt wait

    // hoist loop-invariant bilinear params for this lane's pixel column
    const float w0 = sW[pi*4+0], w1 = sW[pi*4+1], w2 = sW[pi*4+2], w3 = sW[pi*4+3];
    const int   a0 = sA[pi*4+0], a1 = sA[pi*4+1], a2 = sA[pi*4+2], a3 = sA[pi*4+3];

    // cooperative 33-channel bilinear gather into B panel (rows 33..35 = 0)
    #pragma unroll 2
    for (int kk = 0; kk < 18; ++kk) {
        int k = kk * 2 + hi;               // each (k,pi) covered exactly once
        float r = 0.f;
        if (k < 33) {
            const float* sc = skinColor + k * 65536;   // L2-resident channel
            r = w0 * sc[a0] + w1 * sc[a1] + w2 * sc[a2] + w3 * sc[a3];
        }
        sB[k * 16 + pi] = r;
    }
    __syncthreads();

    // preload all 9 A and B fragments, then run the WMMA chain back-to-back
    v2f afr[9], bfr[9];
    #pragma unroll
    for (int s = 0; s < 9; ++s) {
        int k0 = 4 * s + 2 * hi;
        afr[s].x = sPA[pi * 36 + k0];      // unconditional: rows 3..15 are zero
        afr[s].y = sPA[pi * 36 + k0 + 1];
        bfr[s].x = sB[k0 * 16 + pi];
        bfr[s].y = sB[(k0 + 1) * 16 + pi];
    }
    v8f acc = {};
    #pragma unroll
    for (int s = 0; s < 9; ++s) {
        acc = __builtin_amdgcn_wmma_f32_16x16x4_f32(
                  /*neg_a=*/false, afr[s], /*neg_b=*/false, bfr[s],
                  /*c_mod=*/(short)0, acc, /*reuse_a=*/false, /*reuse_b=*/false);
    }

    // epilogue: lanes 0..15 hold C rows M=0..2 in acc[0..2] for pixel N=lane
    if (lane < 16) {
        float d0 = sh_v * acc[0];
        float d1 = sh_v * acc[1];
        float d2 = sh_v * acc[2];
        float LC0 = sMeta[12], LC1 = sMeta[13], LC2 = sMeta[14];
        float sp0 = sp_v * LC0, sp1 = sp_v * LC1, sp2 = sp_v * LC2;
        float r0 = d0 + sp0, r1 = d1 + sp1, r2 = d2 + sp2;
        const float* Mp = sMeta + 15;
        float g0 = fmaxf(Mp[0]*r0 + Mp[1]*r1 + Mp[2]*r2, 0.f);
        float g1 = fmaxf(Mp[3]*r0 + Mp[4]*r1 + Mp[5]*r2, 0.f);
        float g2 = fmaxf(Mp[6]*r0 + Mp[7]*r1 + Mp[8]*r2, 0.f);

        int p     = pbase + lane;
        int base1 = n * HW + p;
        int base3 = n * 3 * HW + p;
        out[OFF_RGB  + base3]          = g0;
        out[OFF_RGB  + base3 + HW]     = g1;
        out[OFF_RGB  + base3 + 2*HW]   = g2;
        out[OFF_RAW  + base3]          = r0;
        out[OFF_RAW  + base3 + HW]     = r1;
        out[OFF_RAW  + base3 + 2*HW]   = r2;
        out[OFF_SPEC + base3]          = sp0;
        out[OFF_SPEC + base3 + HW]     = sp1;
        out[OFF_SPEC + base3 + 2*HW]   = sp2;
        out[OFF_SHADE + base1] = sh_v;
        out[OFF_BLOOD + base1] = bl_v;
        out[OFF_MEL   + base1] = m_v;
    }
}

extern "C" void kernel_launch(void* const* d_in, const int* in_sizes, int n_in,
                              void* d_out, int out_size, void* d_ws, size_t ws_size,
                              hipStream_t stream)
{
    const float* lighting      = (const float*)d_in[0];
    const float* mel           = (const float*)d_in[1];
    const float* blood         = (const float*)d_in[2];
    const float* shade         = (const float*)d_in[3];
    const float* spec          = (const float*)d_in[4];
    const float* illA          = (const float*)d_in[5];
    const float* illF          = (const float*)d_in[6];
    const float* pcaMeans      = (const float*)d_in[7];
    const float* pcaComponents = (const float*)d_in[8];
    const float* skinColor     = (const float*)d_in[9];
    const float* tmatrix       = (const float*)d_in[10];
    const float* s0            = (const float*)d_in[11];
    const float* s1            = (const float*)d_in[12];
    const float* s2            = (const float*)d_in[13];
    const float* t2rgb         = (const float*)d_in[14];

    float* params = (float*)d_ws;              // 256 * 128 floats = 128 KB
    float* out    = (float*)d_out;

    bpn_prep<<<dim3(8), dim3(32), 0, stream>>>(
        lighting, illA, illF, pcaMeans, pcaComponents, tmatrix,
        s0, s1, s2, t2rgb, params, out);

    bpn_pixel<<<dim3(NSAMP * 256), dim3(32), 0, stream>>>(
        mel, blood, shade, spec, skinColor, params, out);
}